// GPUEventSimulator_76553497084686
// MI455X (gfx1250) — compile-verified
//
#include <hip/hip_runtime.h>

// Per-pixel DVS event-count simulation for MI455X (gfx1250, wave32).
// Memory-bound elementwise recurrence: one thread per (h,w) pixel, loops over
// all B=4 batches and T=16 timesteps. No matmul structure -> no WMMA; the
// gfx1250 win is the data-movement path (NT-hinted global_load_b128 of each
// pixel's contiguous 256B image chunk, scale_offset coalesced state loads,
// uniform math scalarized to SALU) plus a single-division reformulation of the
// event-count step (n_pos/n_neg are mutually exclusive, so one IEEE divide
// reproduces the reference bit-exactly for dt >= 0).

typedef __attribute__((ext_vector_type(4))) float v4f;

namespace {
constexpr int Bc  = 4;
constexpr int Hc  = 480;
constexpr int Wc  = 640;
constexpr int Tc  = 16;
constexpr int HWc = Hc * Wc;
constexpr float GOLDf = 0.6180339887f;  // f32-rounded, matching JAX weak-type promotion
}

__global__ __launch_bounds__(256)
void GPUEventSimulator_76553497084686_kernel(
    const float* __restrict__ log_images,        // (H, W, B*T)
    const float* __restrict__ image_ts,          // (B, T)
    const int*   __restrict__ first_times,       // (B,)
    const float* __restrict__ thresholds,        // (2, B, H, W)
    const float* __restrict__ log_states,        // (B, H, W)
    const float* __restrict__ timestamps,        // (B, H, W)
    const float* __restrict__ prev_image_ts,     // (B,)
    const float* __restrict__ rng_states,        // (B, H, W)
    const float* __restrict__ leak_rates,        // (B,)
    const float* __restrict__ shot_rates,        // (B,)
    const float* __restrict__ threshold_mus,     // (B,)
    const float* __restrict__ refractory_periods,// (B,)
    float* __restrict__ out)                     // (B, H, W)
{
    const int p = blockIdx.x * blockDim.x + threadIdx.x;
    if (p >= HWc) return;

    // Pixel p's image data: 64 contiguous floats = 16 float4s.
    const v4f* __restrict__ img4 = (const v4f*)log_images;

    #pragma unroll
    for (int b = 0; b < Bc; ++b) {
        // --- uniform per-batch scalars (scalarized -> SGPRs) ---
        const bool  ft     = first_times[b] > 0;
        const float refr   = refractory_periods[b];
        const float leakmu = leak_rates[b] * threshold_mus[b];
        const float shot   = shot_rates[b];

        // --- per-lane state (coalesced in p) ---
        const float Cp  = thresholds[(size_t)b * HWc + p];
        const float Cn  = thresholds[(size_t)(Bc + b) * HWc + p];
        const float rng = rng_states[(size_t)b * HWc + p];

        // Stream-once image chunk: 4x global_load_b128 with NT hint.
        v4f im[4];
        #pragma unroll
        for (int g = 0; g < 4; ++g)
            im[g] = __builtin_nontemporal_load(&img4[(size_t)p * (Bc * Tc / 4) + b * 4 + g]);

        const float prev0 = ft ? image_ts[b * Tc + 0] : prev_image_ts[b];
        float state   = ft ? im[0].x : log_states[(size_t)b * HWc + p];
        float last_ts = ft ? (prev0 - refr) : timestamps[(size_t)b * HWc + p];

        float counts = 0.0f;
        float tprev  = prev0;

        #pragma unroll
        for (int t = 0; t < Tc; ++t) {
            const float img = im[t >> 2][t & 3];
            const float ts  = image_ts[b * Tc + t];          // uniform (SGPR)
            const float dt  = ts - tprev;                    // uniform
            const float state_l = state - leakmu * dt;
            const float diff = img - state_l;

            // n_pos/n_neg are mutually exclusive: only the branch matching
            // sign(diff) can be nonzero, so one exact IEEE divide suffices
            // and reproduces the reference bit-for-bit (cap >= 1 here since
            // dt > 0, so the inactive branch's min(0, cap) == 0).
            const bool  pos = diff > 0.0f;
            const float C   = pos ? Cp : Cn;
            float q = floorf(fabsf(diff) / C);               // exact IEEE divide

            const float cap = floorf(dt / refr) + 1.0f;      // uniform
            const bool  can = (ts - last_ts) >= refr;
            q = can ? fminf(q, cap) : 0.0f;                  // == n_pos + n_neg

            const bool fired = q > 0.0f;
            const float qs = pos ? q : -q;                   // +n_pos or -n_neg
            state   = fmaf(qs, C, state_l);                  // state_l + n_pos*Cp - n_neg*Cn
            last_ts = fired ? ts : last_ts;

            // r = mod(rng + t*GOLD, 1.0); operands >= 0 so mod == x - floor(x)
            const float x = rng + (float)t * GOLDf;          // t*GOLD folds to constant
            const float r = x - floorf(x);
            const float shot_ev = ((r < shot * dt) && !fired && can) ? 1.0f : 0.0f;

            counts += q + shot_ev;
            tprev = ts;
        }

        __builtin_nontemporal_store(counts, &out[(size_t)b * HWc + p]);
    }
}

extern "C" void kernel_launch(void* const* d_in, const int* in_sizes, int n_in,
                              void* d_out, int out_size, void* d_ws, size_t ws_size,
                              hipStream_t stream) {
    (void)in_sizes; (void)n_in; (void)out_size; (void)d_ws; (void)ws_size;

    // setup_inputs() order:
    // 0 log_images, 1 video_len (unused), 2 image_ts, 3 first_times,
    // 4 thresholds, 5 log_states, 6 timestamps, 7 prev_image_ts,
    // 8 rng_states, 9 leak_rates, 10 shot_rates, 11 threshold_mus,
    // 12 refractory_periods
    const float* log_images    = (const float*)d_in[0];
    const float* image_ts      = (const float*)d_in[2];
    const int*   first_times   = (const int*)  d_in[3];
    const float* thresholds    = (const float*)d_in[4];
    const float* log_states    = (const float*)d_in[5];
    const float* timestamps    = (const float*)d_in[6];
    const float* prev_image_ts = (const float*)d_in[7];
    const float* rng_states    = (const float*)d_in[8];
    const float* leak_rates    = (const float*)d_in[9];
    const float* shot_rates    = (const float*)d_in[10];
    const float* threshold_mus = (const float*)d_in[11];
    const float* refractory    = (const float*)d_in[12];
    float* out = (float*)d_out;

    const int threads = 256;                            // 8 wave32s per block
    const int blocks  = (HWc + threads - 1) / threads;  // 1200 blocks

    GPUEventSimulator_76553497084686_kernel<<<blocks, threads, 0, stream>>>(
        log_images, image_ts, first_times, thresholds, log_states, timestamps,
        prev_image_ts, rng_states, leak_rates, shot_rates, threshold_mus,
        refractory, out);
}